// SinkhornDistance_53609781789177
// MI455X (gfx1250) — compile-verified
//
#include <hip/hip_runtime.h>
#include <hip/hip_bf16.h>
#include <math.h>

#define EPS_F     0.1f
#define INV_EPS   10.0f
#define THRESH_F  0.1f
#define MAX_ITER  100
#define NB        4
#define P         2048
#define KDIM      64
#define ROW_CHUNKS 8
#define ROWS_PER_CHUNK (P / ROW_CHUNKS)   // 256

typedef __attribute__((ext_vector_type(2))) float v2f;
typedef __attribute__((ext_vector_type(4))) float v4f;
typedef __attribute__((ext_vector_type(8))) float v8f;

// Online / pairwise logsumexp merge: (m,s) <- merge((m,s),(om,os))
__device__ __forceinline__ void lse_merge(float& m, float& s, float om, float os) {
    float nm = fmaxf(m, om);
    s = s * __expf(m - nm) + os * __expf(om - nm);
    m = nm;
}

// ---------------------------------------------------------------------------
// 0) init: zero u, v, done
// ---------------------------------------------------------------------------
__global__ void sk_init(float* u, float* v, int* done) {
    int t = blockIdx.x * blockDim.x + threadIdx.x;
    for (int i = t; i < NB * P; i += gridDim.x * blockDim.x) {
        u[i] = 0.0f;
        v[i] = 0.0f;
    }
    if (t == 0) *done = 0;
}

// ---------------------------------------------------------------------------
// 1) row norms: xsq[b,i] = sum_k x^2, ysq[b,j] = sum_k y^2
// ---------------------------------------------------------------------------
__global__ void sk_rownorm(const float* __restrict__ x, const float* __restrict__ y,
                           float* __restrict__ xsq, float* __restrict__ ysq) {
    int idx = blockIdx.x * blockDim.x + threadIdx.x;   // 0 .. NB*P-1
    if (idx >= NB * P) return;
    const float* px = x + (long)idx * KDIM;
    const float* py = y + (long)idx * KDIM;
    float sx = 0.0f, sy = 0.0f;
    #pragma unroll
    for (int k = 0; k < KDIM; k += 4) {
        v4f xv = *(const v4f*)(px + k);
        v4f yv = *(const v4f*)(py + k);
        #pragma unroll
        for (int q = 0; q < 4; ++q) { sx += xv[q] * xv[q]; sy += yv[q] * yv[q]; }
    }
    xsq[idx] = sx;
    ysq[idx] = sy;
}

// ---------------------------------------------------------------------------
// 2) Cost matrix via f32 WMMA:  C = xsq_i + ysq_j - 2 * (x . y^T)
//    One wave computes a 16(M) x 64(N) tile; A fragment reused for 4 WMMAs.
//    A-frag (16x4 f32): lanes 0-15 -> M=row, K={k0,k0+1}; lanes 16-31 K={k0+2,k0+3}
//    B-frag (4x16 f32): mirrored (lane-half splits K), so both are float2 loads.
//    D (16x16 f32): VGPR r, lane l -> M = r + 8*(l>>4), N = l&15.
// ---------------------------------------------------------------------------
__global__ void sk_cost_wmma(const float* __restrict__ x, const float* __restrict__ y,
                             const float* __restrict__ xsq, const float* __restrict__ ysq,
                             float* __restrict__ C) {
    const int lane = threadIdx.x & 31;
    const int wave = threadIdx.x >> 5;
    const long gwave = (long)blockIdx.x * (blockDim.x >> 5) + wave;
    // per batch: 128 M-tiles * 32 N64-tiles = 4096 wave-tiles
    const int b  = (int)(gwave >> 12);
    const int t  = (int)(gwave & 4095);
    const int mt = t >> 5;          // 0..127
    const int nt = t & 31;          // 0..31
    const int Mbase = mt * 16;
    const int Nbase = nt * 64;
    const int half = lane >> 4;
    const int row  = lane & 15;

    const float* xb = x + (long)b * P * KDIM;
    const float* yb = y + (long)b * P * KDIM;

    v8f acc0 = {}, acc1 = {}, acc2 = {}, acc3 = {};
    #pragma unroll
    for (int k0 = 0; k0 < KDIM; k0 += 4) {
        const int ko = k0 + 2 * half;
        v2f a  = *(const v2f*)(xb + (long)(Mbase + row) * KDIM + ko);
        v2f b0 = *(const v2f*)(yb + (long)(Nbase +  0 + row) * KDIM + ko);
        v2f b1 = *(const v2f*)(yb + (long)(Nbase + 16 + row) * KDIM + ko);
        v2f b2 = *(const v2f*)(yb + (long)(Nbase + 32 + row) * KDIM + ko);
        v2f b3 = *(const v2f*)(yb + (long)(Nbase + 48 + row) * KDIM + ko);
        acc0 = __builtin_amdgcn_wmma_f32_16x16x4_f32(false, a, false, b0, (short)0, acc0, false, false);
        acc1 = __builtin_amdgcn_wmma_f32_16x16x4_f32(false, a, false, b1, (short)0, acc1, false, false);
        acc2 = __builtin_amdgcn_wmma_f32_16x16x4_f32(false, a, false, b2, (short)0, acc2, false, false);
        acc3 = __builtin_amdgcn_wmma_f32_16x16x4_f32(false, a, false, b3, (short)0, acc3, false, false);
    }

    const float ys0 = ysq[(long)b * P + Nbase +  0 + row];
    const float ys1 = ysq[(long)b * P + Nbase + 16 + row];
    const float ys2 = ysq[(long)b * P + Nbase + 32 + row];
    const float ys3 = ysq[(long)b * P + Nbase + 48 + row];

    #pragma unroll
    for (int r = 0; r < 8; ++r) {
        const int M = Mbase + r + 8 * half;
        const float xs = xsq[(long)b * P + M];
        float* Crow = C + ((long)b * P + M) * P;
        Crow[Nbase +  0 + row] = xs + ys0 - 2.0f * acc0[r];
        Crow[Nbase + 16 + row] = xs + ys1 - 2.0f * acc1[r];
        Crow[Nbase + 32 + row] = xs + ys2 - 2.0f * acc2[r];
        Crow[Nbase + 48 + row] = xs + ys3 - 2.0f * acc3[r];
    }
}

// ---------------------------------------------------------------------------
// 3) u update: one wave per row i, b128 streaming loads.
//    In-place (u[i] depends only on u[i], v, C row). Writes |du| for convergence.
// ---------------------------------------------------------------------------
__global__ void sk_u_update(const float* __restrict__ C, float* __restrict__ u,
                            const float* __restrict__ v, float* __restrict__ du,
                            const int* __restrict__ done) {
    if (*done) return;
    const float LOG_MU = __logf(1.0f / (float)P + 1e-8f);
    const int lane = threadIdx.x & 31;
    const int wave = threadIdx.x >> 5;
    const long grow = (long)blockIdx.x * (blockDim.x >> 5) + wave;  // 0..NB*P-1
    const int b = (int)(grow >> 11);
    const int i = (int)(grow & (P - 1));
    const float* Crow = C + ((long)b * P + i) * P;
    const float* vb = v + (long)b * P;
    const float ui = u[(long)b * P + i];

    float m = -INFINITY, s = 0.0f;
    for (int j = lane * 4; j < P; j += 128) {
        v4f c4 = *(const v4f*)(Crow + j);
        v4f vv = *(const v4f*)(vb + j);
        #pragma unroll
        for (int q = 0; q < 4; ++q) {
            float val = (-c4[q] + ui + vv[q]) * INV_EPS;
            lse_merge(m, s, val, 1.0f);
        }
    }
    // wave32 butterfly reduce of (m, s)
    #pragma unroll
    for (int off = 16; off > 0; off >>= 1) {
        float om = __shfl_xor(m, off, 32);
        float os = __shfl_xor(s, off, 32);
        lse_merge(m, s, om, os);
    }
    if (lane == 0) {
        float lse = m + __logf(s);
        float un = EPS_F * (LOG_MU - lse) + ui;
        u[(long)b * P + i] = un;
        du[(long)b * P + i] = fabsf(un - ui);
    }
}

// ---------------------------------------------------------------------------
// 4a) v partial: column logsumexp split into ROW_CHUNKS row-chunks for
//     parallelism (256 workgroups instead of 32). Coalesced column loads.
// ---------------------------------------------------------------------------
__global__ void sk_v_partial(const float* __restrict__ C, const float* __restrict__ u,
                             const float* __restrict__ v, float* __restrict__ pm,
                             float* __restrict__ ps, const int* __restrict__ done) {
    if (*done) return;
    const int j     = blockIdx.x * blockDim.x + threadIdx.x;  // 0..P-1
    const int chunk = blockIdx.y;
    const int b     = blockIdx.z;
    const float* Cb = C + (long)b * P * P;
    const float* ub = u + (long)b * P;
    const float vj = v[(long)b * P + j];

    float m = -INFINITY, s = 0.0f;
    const int i0 = chunk * ROWS_PER_CHUNK;
    for (int i = i0; i < i0 + ROWS_PER_CHUNK; ++i) {
        float val = (-Cb[(long)i * P + j] + ub[i] + vj) * INV_EPS;
        lse_merge(m, s, val, 1.0f);
    }
    const long pidx = ((long)b * ROW_CHUNKS + chunk) * P + j;
    pm[pidx] = m;
    ps[pidx] = s;
}

// 4b) v combine: merge the ROW_CHUNKS partials per column, update v in place.
__global__ void sk_v_combine(const float* __restrict__ pm, const float* __restrict__ ps,
                             float* __restrict__ v, const int* __restrict__ done) {
    if (*done) return;
    const float LOG_NU = __logf(1.0f / (float)P + 1e-8f);
    const int t = blockIdx.x * blockDim.x + threadIdx.x;  // 0..NB*P-1
    const int b = t >> 11;
    const int j = t & (P - 1);
    float m = -INFINITY, s = 0.0f;
    #pragma unroll
    for (int c = 0; c < ROW_CHUNKS; ++c) {
        const long pidx = ((long)b * ROW_CHUNKS + c) * P + j;
        lse_merge(m, s, pm[pidx], ps[pidx]);
    }
    float lse = m + __logf(s);
    v[(long)b * P + j] = EPS_F * (LOG_NU - lse) + v[(long)b * P + j];
}

// ---------------------------------------------------------------------------
// 5) convergence: err = mean_b sum_i |du| = (sum all du)/NB ; sticky done flag.
//    Single block, fixed-order reduction (deterministic).
// ---------------------------------------------------------------------------
__global__ void sk_check_done(const float* __restrict__ du, int* done) {
    if (*done) return;  // uniform branch
    __shared__ float sm[256];
    float s = 0.0f;
    for (int t = threadIdx.x; t < NB * P; t += 256) s += du[t];
    sm[threadIdx.x] = s;
    __syncthreads();
    for (int o = 128; o > 0; o >>= 1) {
        if (threadIdx.x < o) sm[threadIdx.x] += sm[threadIdx.x + o];
        __syncthreads();
    }
    if (threadIdx.x == 0 && (sm[0] * (1.0f / (float)NB)) < THRESH_F) *done = 1;
}

// ---------------------------------------------------------------------------
// 6) pi = exp(M) (b128 in/out, 4 elems/thread), per-block partial sums of pi*C
// ---------------------------------------------------------------------------
__global__ void sk_transport(const float* __restrict__ C, const float* __restrict__ u,
                             const float* __restrict__ v, float* __restrict__ pi,
                             float* __restrict__ partial) {
    const int b = blockIdx.y;
    const long e = ((long)blockIdx.x * blockDim.x + threadIdx.x) * 4;  // < P*P
    const int i = (int)(e >> 11);
    const int j = (int)(e & (P - 1));
    v4f c4 = *(const v4f*)(C + ((long)b << 22) + e);
    const float ui = u[(long)b * P + i];
    v4f vv = *(const v4f*)(v + (long)b * P + j);
    v4f p4;
    float acc = 0.0f;
    #pragma unroll
    for (int q = 0; q < 4; ++q) {
        float p = __expf((-c4[q] + ui + vv[q]) * INV_EPS);
        p4[q] = p;
        acc += p * c4[q];
    }
    *(v4f*)(pi + ((long)b << 22) + e) = p4;

    __shared__ float sm[256];
    sm[threadIdx.x] = acc;
    __syncthreads();
    for (int o = 128; o > 0; o >>= 1) {
        if (threadIdx.x < o) sm[threadIdx.x] += sm[threadIdx.x + o];
        __syncthreads();
    }
    if (threadIdx.x == 0) partial[(long)b * gridDim.x + blockIdx.x] = sm[0];
}

__global__ void sk_cost_reduce(const float* __restrict__ partial, float* __restrict__ cost,
                               int nparts) {
    const int b = blockIdx.x;
    __shared__ float sm[256];
    float s = 0.0f;
    for (int t = threadIdx.x; t < nparts; t += 256) s += partial[(long)b * nparts + t];
    sm[threadIdx.x] = s;
    __syncthreads();
    for (int o = 128; o > 0; o >>= 1) {
        if (threadIdx.x < o) sm[threadIdx.x] += sm[threadIdx.x + o];
        __syncthreads();
    }
    if (threadIdx.x == 0) cost[b] = sm[0];
}

// ---------------------------------------------------------------------------
extern "C" void kernel_launch(void* const* d_in, const int* in_sizes, int n_in,
                              void* d_out, int out_size, void* d_ws, size_t ws_size,
                              hipStream_t stream) {
    const float* x = (const float*)d_in[0];
    const float* y = (const float*)d_in[1];
    float* out = (float*)d_out;

    // Output layout: [cost (NB)] [pi (NB*P*P)] [C (NB*P*P)]
    float* cost = out;
    float* pi   = out + NB;
    float* Cm   = out + NB + (long)NB * P * P;

    // Workspace layout (floats)
    float* ws   = (float*)d_ws;
    float* xsq  = ws;                          // NB*P
    float* ysq  = xsq + NB * P;                // NB*P
    float* u    = ysq + NB * P;                // NB*P
    float* v    = u   + NB * P;                // NB*P
    float* du   = v   + NB * P;                // NB*P
    float* pm   = du  + NB * P;                // NB*ROW_CHUNKS*P
    float* psum = pm  + (long)NB * ROW_CHUNKS * P;   // NB*ROW_CHUNKS*P
    float* part = psum + (long)NB * ROW_CHUNKS * P;  // NB * (P*P/1024)
    int*   done = (int*)(part + (long)NB * (P * P / 1024));

    // 0) init
    sk_init<<<32, 256, 0, stream>>>(u, v, done);

    // 1) row norms
    sk_rownorm<<<(NB * P + 255) / 256, 256, 0, stream>>>(x, y, xsq, ysq);

    // 2) cost matrix via WMMA: NB*4096 wave-tiles, 8 waves per 256-thread block
    sk_cost_wmma<<<(NB * 4096) / 8, 256, 0, stream>>>(x, y, xsq, ysq, Cm);

    // 3-5) Sinkhorn iterations (fixed count; kernels freeze via device flag)
    for (int it = 0; it < MAX_ITER; ++it) {
        sk_u_update<<<(NB * P) / 8, 256, 0, stream>>>(Cm, u, v, du, done);
        sk_v_partial<<<dim3(P / 256, ROW_CHUNKS, NB), 256, 0, stream>>>(Cm, u, v, pm, psum, done);
        sk_v_combine<<<(NB * P) / 256, 256, 0, stream>>>(pm, psum, v, done);
        sk_check_done<<<1, 256, 0, stream>>>(du, done);
    }

    // 6) transport plan + cost
    const int nparts = P * P / 1024;
    sk_transport<<<dim3(nparts, NB), 256, 0, stream>>>(Cm, u, v, pi, part);
    sk_cost_reduce<<<NB, 256, 0, stream>>>(part, cost, nparts);

    (void)in_sizes; (void)n_in; (void)out_size; (void)ws_size;
}